// HGT_22608707846795
// MI455X (gfx1250) — compile-verified
//
#include <hip/hip_runtime.h>
#include <hip/hip_bf16.h>
#include <math.h>

// ---------------------------------------------------------------------------
// HGT forward for gfx1250 (MI455X). Dense GEMMs use V_WMMA_F32_16X16X4_F32
// (fp32 matrix cores, wave32) with LDS-staged, double-buffered B tiles and
// 32x64 C strips per wave. Edge attention uses L2 float atomics.
// ---------------------------------------------------------------------------

typedef float v2f __attribute__((ext_vector_type(2)));
typedef float v8f __attribute__((ext_vector_type(8)));

#define NINST 30000
#define NVAR  30000
#define NEDGE 200000
#define NGRAPH 64
#define SUMHID 368
#define KCHUNK 16

__device__ __forceinline__ float gelu_erf(float x) {
  return 0.5f * x * (1.0f + erff(x * 0.70710678118654752440f));
}

__device__ __forceinline__ v8f wmma4(v2f a, v2f b, v8f c) {
  return __builtin_amdgcn_wmma_f32_16x16x4_f32(false, a, false, b, (short)0, c,
                                               false, false);
}

// ---------------------------------------------------------------------------
// Main WMMA fp32 GEMM: C[M, n0:n0+64] = act(A @ W + bias) for n0 = blockIdx.x*64.
// Preconditions (hold for every GEMM in this net): M%16==0, K%16==0, and the
// launched column range is fully in-bounds (host splits off the N%64 tail).
// block = 256 threads = 8 waves; wave w computes TWO m-tiles (32 rows) over a
// 64-column strip: B fragments from LDS are reused across both m-tiles
// (1.0 b32-loads per WMMA). B chunk (16K x 64N) is staged in LDS once per
// block (8x reduction in W traffic), double-buffered, one barrier per chunk.
// All control flow is wave-uniform => EXEC all-ones at every WMMA; A-row
// indices are clamped and stores guarded so no wave skips a barrier.
// ---------------------------------------------------------------------------
__global__ __launch_bounds__(256) void gemm_wmma_main(
    const float* __restrict__ A, const float* __restrict__ W,
    const float* __restrict__ bias, float* __restrict__ C,
    int M, int K, int N, int act) {
  __shared__ float Bs[2][KCHUNK][64];

  const int lane = threadIdx.x & 31;
  const int wave = threadIdx.x >> 5;
  const int m0 = (blockIdx.y * 8 + wave) << 5;   // 32 rows per wave
  const bool valid0 = (m0 +  0) < M;             // wave-uniform store guards
  const bool valid1 = (m0 + 16) < M;
  const int n0 = blockIdx.x << 6;
  const int row = lane & 15;                 // M row (A frag) / N col (B,C frags)
  const int kk = (lane >> 4) << 1;           // K pair: 0 (lanes 0-15) or 2 (16-31)

  // clamped A row indices (loads always in-bounds; bogus rows never stored)
  const int r0 = min(m0 + row, M - 1);
  const int r1 = min(m0 + 16 + row, M - 1);
  const float* Arow0 = A + (size_t)r0 * K;
  const float* Arow1 = A + (size_t)r1 * K;

  // cooperative B loader: thread t -> chunk row t>>4, 4 consecutive cols
  const int kr = threadIdx.x >> 4;
  const int cb = (threadIdx.x & 15) << 2;

  v8f acc00 = {}, acc01 = {}, acc02 = {}, acc03 = {};   // m-tile 0, n-subtiles 0..3
  v8f acc10 = {}, acc11 = {}, acc12 = {}, acc13 = {};   // m-tile 1

  // preload chunk 0 into buffer 0
  *(float4*)&Bs[0][kr][cb] = *(const float4*)(W + (size_t)kr * N + n0 + cb);
  __syncthreads();

  const int nchunks = K / KCHUNK;
  for (int c = 0; c < nchunks; ++c) {
    const int buf = c & 1;
    if (c + 1 < nchunks) {                   // prefetch next chunk into other buf
      *(float4*)&Bs[buf ^ 1][kr][cb] =
          *(const float4*)(W + (size_t)((c + 1) * KCHUNK + kr) * N + n0 + cb);
    }
    const int k0 = c * KCHUNK;
    if (k0 + 2 * KCHUNK < K) {
      __builtin_prefetch(Arow0 + k0 + 2 * KCHUNK, 0, 1);
      __builtin_prefetch(Arow1 + k0 + 2 * KCHUNK, 0, 1);
    }
#pragma unroll
    for (int ks = 0; ks < KCHUNK; ks += 4) {
      v2f a0, a1;
      a0.x = Arow0[k0 + ks + kk];
      a0.y = Arow0[k0 + ks + kk + 1];
      a1.x = Arow1[k0 + ks + kk];
      a1.y = Arow1[k0 + ks + kk + 1];
      v2f b0, b1, b2, b3;
      b0.x = Bs[buf][ks + kk][ 0 + row]; b0.y = Bs[buf][ks + kk + 1][ 0 + row];
      b1.x = Bs[buf][ks + kk][16 + row]; b1.y = Bs[buf][ks + kk + 1][16 + row];
      b2.x = Bs[buf][ks + kk][32 + row]; b2.y = Bs[buf][ks + kk + 1][32 + row];
      b3.x = Bs[buf][ks + kk][48 + row]; b3.y = Bs[buf][ks + kk + 1][48 + row];
      acc00 = wmma4(a0, b0, acc00);
      acc01 = wmma4(a0, b1, acc01);
      acc02 = wmma4(a0, b2, acc02);
      acc03 = wmma4(a0, b3, acc03);
      acc10 = wmma4(a1, b0, acc10);
      acc11 = wmma4(a1, b1, acc11);
      acc12 = wmma4(a1, b2, acc12);
      acc13 = wmma4(a1, b3, acc13);
    }
    __syncthreads();
  }

  // C/D layout: VGPR r -> row mbase + r + 8*(lane>=16), col n0 + 16*j + (lane&15)
  const int rbase = (lane >> 4) << 3;
  v8f accs0[4] = {acc00, acc01, acc02, acc03};
  v8f accs1[4] = {acc10, acc11, acc12, acc13};
#pragma unroll
  for (int j = 0; j < 4; ++j) {
    const int col = n0 + j * 16 + row;
    const float bv = bias ? bias[col] : 0.0f;
    if (valid0) {
#pragma unroll
      for (int r = 0; r < 8; ++r) {
        float val = accs0[j][r] + bv;
        if (act == 1) val = gelu_erf(val);
        C[(size_t)(m0 + rbase + r) * N + col] = val;
      }
    }
    if (valid1) {
#pragma unroll
      for (int r = 0; r < 8; ++r) {
        float val = accs1[j][r] + bv;
        if (act == 1) val = gelu_erf(val);
        C[(size_t)(m0 + 16 + rbase + r) * N + col] = val;
      }
    }
  }
}

// ---------------------------------------------------------------------------
// Tail WMMA GEMM: one 16x16 tile per wave at columns n_base + blockIdx.x*16.
// Direct global loads; K%4==0. No barriers, so wave-uniform early-exit is fine.
// ---------------------------------------------------------------------------
__global__ __launch_bounds__(256) void gemm_wmma_tail(
    const float* __restrict__ A, const float* __restrict__ W,
    const float* __restrict__ bias, float* __restrict__ C,
    int M, int K, int N, int act, int n_base) {
  const int lane = threadIdx.x & 31;
  const int wave = threadIdx.x >> 5;
  const int mtile = blockIdx.y * 8 + wave;
  if (mtile * 16 >= M) return;              // wave-uniform
  const int m0 = mtile << 4;
  const int n0 = n_base + (blockIdx.x << 4);
  const int row = lane & 15;
  const int kk = (lane >> 4) << 1;

  v8f acc = {};
  const float* Arow = A + (size_t)(m0 + row) * K;
  for (int k0 = 0; k0 < K; k0 += 4) {
    v2f a, b;
    a.x = Arow[k0 + kk];
    a.y = Arow[k0 + kk + 1];
    b.x = W[(size_t)(k0 + kk) * N + n0 + row];
    b.y = W[(size_t)(k0 + kk + 1) * N + n0 + row];
    acc = wmma4(a, b, acc);
  }
  const int rbase = (lane >> 4) << 3;
  const int col = n0 + row;
  const float bv = bias ? bias[col] : 0.0f;
#pragma unroll
  for (int r = 0; r < 8; ++r) {
    float val = acc[r] + bv;
    if (act == 1) val = gelu_erf(val);
    C[(size_t)(m0 + rbase + r) * N + col] = val;
  }
}

// ---------------------------------------------------------------------------
// Elementwise / utility kernels
// ---------------------------------------------------------------------------
__global__ void fill_f32(float* __restrict__ p, float v, int n) {
  int i = blockIdx.x * blockDim.x + threadIdx.x;
  if (i < n) p[i] = v;
}

__global__ void gelu_inplace(float* __restrict__ p, int n) {
  int i = blockIdx.x * blockDim.x + threadIdx.x;
  if (i < n) p[i] = gelu_erf(p[i]);
}

// h = gelu(h * (g*bnscale) + be), h is [G, C]
__global__ void bn_gelu(float* __restrict__ h, const float* __restrict__ g,
                        const float* __restrict__ be, float bnscale, int n, int C) {
  int i = blockIdx.x * blockDim.x + threadIdx.x;
  if (i >= n) return;
  int c = i % C;
  h[i] = gelu_erf(h[i] * (g[c] * bnscale) + be[c]);
}

// out[n, h*D+e] = sum_d in[n, colOff + h*D + d] * Wr[h][d][e]
__global__ void rel_transform(const float* __restrict__ in, int ldin, int colOff,
                              const float* __restrict__ Wr,
                              float* __restrict__ out, int Nn, int H, int D) {
  const int F = H * D;
  int idx = blockIdx.x * blockDim.x + threadIdx.x;
  if (idx >= Nn * F) return;
  int n = idx / F;
  int f = idx - n * F;
  int h = f / D;
  int e = f - h * D;
  const float* xi = in + (size_t)n * ldin + colOff + h * D;
  const float* wr = Wr + (size_t)h * D * D + e;
  float s = 0.f;
  for (int d = 0; d < D; ++d) s += xi[d] * wr[(size_t)d * D];
  out[idx] = s;
}

// logit[e,h] = scale * p_rel[h] * dot(q[dst[e],h,:], krel[src[e],h,:])
__global__ void edge_logits(const int* __restrict__ src, const int* __restrict__ dst,
                            const float* __restrict__ kqv_dst, int ld_dst, int qOff,
                            const float* __restrict__ krel_src,
                            const float* __restrict__ p_rel, float scale,
                            float* __restrict__ logit, int E, int H, int D) {
  int idx = blockIdx.x * blockDim.x + threadIdx.x;
  if (idx >= E * H) return;
  int e = idx / H;
  int h = idx - e * H;
  const float* q = kqv_dst + (size_t)dst[e] * ld_dst + qOff + h * D;
  const float* k = krel_src + (size_t)src[e] * (H * D) + h * D;
  float s = 0.f;
  for (int d = 0; d < D; ++d) s += q[d] * k[d];
  logit[idx] = s * p_rel[h] * scale;
}

__device__ __forceinline__ void atomicMaxFloat(float* addr, float v) {
  if (v >= 0.0f) atomicMax((int*)addr, __float_as_int(v));
  else           atomicMin((unsigned int*)addr, __float_as_uint(v));
}

__global__ void seg_max(const int* __restrict__ dst, const float* __restrict__ logit,
                        float* __restrict__ segmax, int E, int H) {
  int idx = blockIdx.x * blockDim.x + threadIdx.x;
  if (idx >= E * H) return;
  int e = idx / H;
  int h = idx - e * H;
  atomicMaxFloat(&segmax[(size_t)dst[e] * H + h], logit[idx]);
}

__global__ void seg_max_fix(float* __restrict__ segmax, int n) {
  int i = blockIdx.x * blockDim.x + threadIdx.x;
  if (i >= n) return;
  float v = segmax[i];
  if (!isfinite(v)) segmax[i] = 0.0f;
}

// ev = exp(logit - segmax[dst]); logit <- ev; segsum[dst] += ev
__global__ void edge_exp_sum(const int* __restrict__ dst, float* __restrict__ logit,
                             const float* __restrict__ segmax, float* __restrict__ segsum,
                             int E, int H) {
  int idx = blockIdx.x * blockDim.x + threadIdx.x;
  if (idx >= E * H) return;
  int e = idx / H;
  int h = idx - e * H;
  size_t s = (size_t)dst[e] * H + h;
  float ev = expf(logit[idx] - segmax[s]);
  logit[idx] = ev;
  atomicAdd(&segsum[s], ev);
}

// agg[dst, f] += (ev[e,h] / (segsum[dst,h] + 1e-16)) * vrel[src, f]
__global__ void edge_aggregate(const int* __restrict__ src, const int* __restrict__ dst,
                               const float* __restrict__ ev, const float* __restrict__ segsum,
                               const float* __restrict__ vrel, float* __restrict__ agg,
                               int E, int H, int D) {
  const int F = H * D;
  int idx = blockIdx.x * blockDim.x + threadIdx.x;
  if (idx >= E * F) return;
  int e = idx / F;
  int f = idx - e * F;
  int h = f / D;
  int de = dst[e];
  float a = ev[(size_t)e * H + h] / (segsum[(size_t)de * H + h] + 1e-16f);
  atomicAdd(&agg[(size_t)de * F + f], a * vrel[(size_t)src[e] * F + f]);
}

// LayerNorm over last dim F; one wave32 per row; writes x_cur and JK slice.
__global__ void layernorm_dual(const float* __restrict__ x, const float* __restrict__ g,
                               const float* __restrict__ b, float* __restrict__ out_cur,
                               float* __restrict__ out_jk, int ld_jk, int col_off,
                               int Nn, int F) {
  int wave = threadIdx.x >> 5;
  int lane = threadIdx.x & 31;
  int n = blockIdx.x * (blockDim.x >> 5) + wave;
  if (n >= Nn) return;
  const float* xr = x + (size_t)n * F;
  float sum = 0.f, sumsq = 0.f;
  for (int i = lane; i < F; i += 32) {
    float v = xr[i];
    sum += v; sumsq += v * v;
  }
  for (int o = 16; o > 0; o >>= 1) {
    sum   += __shfl_xor(sum, o, 32);
    sumsq += __shfl_xor(sumsq, o, 32);
  }
  float inv = 1.0f / (float)F;
  float mu = sum * inv;
  float var = sumsq * inv - mu * mu;
  float rs = rsqrtf(var + 1e-5f);
  for (int i = lane; i < F; i += 32) {
    float v = (xr[i] - mu) * rs * g[i] + b[i];
    out_cur[(size_t)n * F + i] = v;
    out_jk[(size_t)n * ld_jk + col_off + i] = v;
  }
}

// pooled[batch[n], colOff + c] += x[n, c]
__global__ void pool_sum(const float* __restrict__ x, const int* __restrict__ batch,
                         float* __restrict__ h, int Nn, int C, int colOff, int ldh) {
  int idx = blockIdx.x * blockDim.x + threadIdx.x;
  if (idx >= Nn * C) return;
  int n = idx / C;
  int c = idx - n * C;
  atomicAdd(&h[(size_t)batch[n] * ldh + colOff + c], x[idx]);
}

// out[g] = dot(h[g,:], w) + b[0]
__global__ void final_out(const float* __restrict__ h, const float* __restrict__ w,
                          const float* __restrict__ b, float* __restrict__ out,
                          int Gn, int K) {
  int g = blockIdx.x * blockDim.x + threadIdx.x;
  if (g >= Gn) return;
  float s = 0.f;
  for (int k = 0; k < K; ++k) s += h[(size_t)g * K + k] * w[k];
  out[g] = s + b[0];
}

// ---------------------------------------------------------------------------
// Host orchestration
// ---------------------------------------------------------------------------
static inline void launch_gemm(const float* A, const float* W, const float* bias, float* C,
                               int M, int K, int N, int act, hipStream_t s) {
  const int Nmain = N & ~63;
  if (Nmain > 0) {
    const int gy = ((M + 31) / 32 + 7) / 8;   // 32 rows per wave, 8 waves/block
    gemm_wmma_main<<<dim3(Nmain / 64, gy), 256, 0, s>>>(A, W, bias, C, M, K, N, act);
  }
  const int ntail = (N - Nmain) / 16;
  if (ntail > 0) {
    const int gy = (M / 16 + 7) / 8;
    gemm_wmma_tail<<<dim3(ntail, gy), 256, 0, s>>>(A, W, bias, C, M, K, N, act, Nmain);
  }
}
static inline int blks(int n) { return (n + 255) / 256; }

extern "C" void kernel_launch(void* const* d_in, const int* in_sizes, int n_in,
                              void* d_out, int out_size, void* d_ws, size_t ws_size,
                              hipStream_t stream) {
  (void)in_sizes; (void)n_in; (void)out_size; (void)ws_size;

  const int HID[5]   = {128, 96, 64, 48, 32};
  const int HEADS[5] = {8, 8, 8, 4, 4};
  const int NN[2]    = {NINST, NVAR};
  const int srcT[3]  = {0, 0, 1};   // edge types: (inst,inst),(inst,var),(var,inst)
  const int dstT[3]  = {0, 1, 0};

  // ---- inputs (insertion order for top-level dict) ----
  const float* x_in[2] = {(const float*)d_in[0], (const float*)d_in[1]};
  const int* ei[3] = {(const int*)d_in[2], (const int*)d_in[3], (const int*)d_in[4]};
  const int* batch[2] = {(const int*)d_in[5], (const int*)d_in[6]};

  // ---- params: JAX pytree order (dict keys sorted at every level) ----
  int pi = 7;
  auto nextp = [&]() { return (const float*)d_in[pi++]; };
  struct LayerP {
    const float *k_rel, *kqv_b[2], *kqv_w[2], *ln_b[2], *ln_g[2],
                *out_b[2], *out_w[2], *p_rel, *v_rel;
  } L[5];
  for (int i = 0; i < 5; ++i) {
    L[i].k_rel   = nextp();
    L[i].kqv_b[0] = nextp(); L[i].kqv_b[1] = nextp();
    L[i].kqv_w[0] = nextp(); L[i].kqv_w[1] = nextp();
    L[i].ln_b[0]  = nextp(); L[i].ln_b[1]  = nextp();
    L[i].ln_g[0]  = nextp(); L[i].ln_g[1]  = nextp();
    L[i].out_b[0] = nextp(); L[i].out_b[1] = nextp();
    L[i].out_w[0] = nextp(); L[i].out_w[1] = nextp();
    L[i].p_rel   = nextp();
    L[i].v_rel   = nextp();
  }
  const float *mlp_w[3], *mlp_b[3], *mlp_g[3], *mlp_be[3];
  for (int i = 0; i < 3; ++i) {
    mlp_w[i] = nextp(); mlp_b[i] = nextp(); mlp_g[i] = nextp(); mlp_be[i] = nextp();
  }
  const float* outw = nextp();
  const float* outb = nextp();
  const float* node_b[2] = {nextp(), nextp()};
  const float* node_w[2] = {nextp(), nextp()};
  const float* proj_b[2] = {nextp(), nextp()};
  const float* proj_w[2] = {nextp(), nextp()};

  // ---- workspace bump allocator ----
  size_t off = 0;
  auto alloc = [&](size_t nfloats) {
    float* p = (float*)((char*)d_ws + off);
    off = (off + nfloats * sizeof(float) + 255) & ~(size_t)255;
    return p;
  };
  float* xcur[2]   = {alloc((size_t)NINST * 256), alloc((size_t)NVAR * 256)};
  float* xs[2]     = {alloc((size_t)NINST * SUMHID), alloc((size_t)NVAR * SUMHID)};
  float* kqv[2]    = {alloc((size_t)NINST * 384), alloc((size_t)NVAR * 384)};
  float* krel[3]   = {alloc((size_t)NINST * 128), alloc((size_t)NINST * 128), alloc((size_t)NVAR * 128)};
  float* vrel[3]   = {alloc((size_t)NINST * 128), alloc((size_t)NINST * 128), alloc((size_t)NVAR * 128)};
  float* logits[3] = {alloc((size_t)NEDGE * 8), alloc((size_t)NEDGE * 8), alloc((size_t)NEDGE * 8)};
  float* segmax[2] = {alloc((size_t)NINST * 8), alloc((size_t)NVAR * 8)};
  float* segsum[2] = {alloc((size_t)NINST * 8), alloc((size_t)NVAR * 8)};
  float* agg[2]    = {alloc((size_t)NINST * 128), alloc((size_t)NVAR * 128)};
  float* ybuf[2]   = {alloc((size_t)NINST * 128), alloc((size_t)NVAR * 128)};
  float* xfin[2]   = {alloc((size_t)NINST * 128), alloc((size_t)NVAR * 128)};
  float* hbuf[2]   = {alloc((size_t)NGRAPH * 256), alloc((size_t)NGRAPH * 256)};

  // ---- input projection: xcur = gelu(x @ proj_w + proj_b) ----
  for (int t = 0; t < 2; ++t)
    launch_gemm(x_in[t], proj_w[t], proj_b[t], xcur[t], NN[t], 64, 256, 1, stream);

  // ---- HGT layers ----
  int jkoff = 0;
  int in_dim = 256;
  for (int li = 0; li < 5; ++li) {
    const int F = HID[li], H = HEADS[li], D = F / H;
    const float scale = 1.0f / sqrtf((float)D);

    for (int t = 0; t < 2; ++t)
      launch_gemm(xcur[t], L[li].kqv_w[t], L[li].kqv_b[t], kqv[t], NN[t], in_dim, 3 * F, 0, stream);

    for (int et = 0; et < 3; ++et) {
      const int st = srcT[et], ns = NN[st];
      rel_transform<<<blks(ns * F), 256, 0, stream>>>(
          kqv[st], 3 * F, 0, L[li].k_rel + (size_t)et * H * D * D, krel[et], ns, H, D);
      rel_transform<<<blks(ns * F), 256, 0, stream>>>(
          kqv[st], 3 * F, 2 * F, L[li].v_rel + (size_t)et * H * D * D, vrel[et], ns, H, D);
    }

    for (int et = 0; et < 3; ++et) {
      const int dt = dstT[et];
      edge_logits<<<blks(NEDGE * H), 256, 0, stream>>>(
          ei[et], ei[et] + NEDGE, kqv[dt], 3 * F, F, krel[et],
          L[li].p_rel + (size_t)et * H, scale, logits[et], NEDGE, H, D);
    }

    for (int t = 0; t < 2; ++t) {
      fill_f32<<<blks(NN[t] * H), 256, 0, stream>>>(segmax[t], -INFINITY, NN[t] * H);
      fill_f32<<<blks(NN[t] * H), 256, 0, stream>>>(segsum[t], 0.0f, NN[t] * H);
      fill_f32<<<blks(NN[t] * F), 256, 0, stream>>>(agg[t], 0.0f, NN[t] * F);
    }
    for (int et = 0; et < 3; ++et)
      seg_max<<<blks(NEDGE * H), 256, 0, stream>>>(ei[et] + NEDGE, logits[et], segmax[dstT[et]], NEDGE, H);
    for (int t = 0; t < 2; ++t)
      seg_max_fix<<<blks(NN[t] * H), 256, 0, stream>>>(segmax[t], NN[t] * H);
    for (int et = 0; et < 3; ++et)
      edge_exp_sum<<<blks(NEDGE * H), 256, 0, stream>>>(
          ei[et] + NEDGE, logits[et], segmax[dstT[et]], segsum[dstT[et]], NEDGE, H);
    for (int et = 0; et < 3; ++et)
      edge_aggregate<<<blks(NEDGE * F), 256, 0, stream>>>(
          ei[et], ei[et] + NEDGE, logits[et], segsum[dstT[et]], vrel[et], agg[dstT[et]], NEDGE, H, D);

    for (int t = 0; t < 2; ++t) {
      gelu_inplace<<<blks(NN[t] * F), 256, 0, stream>>>(agg[t], NN[t] * F);
      launch_gemm(agg[t], L[li].out_w[t], L[li].out_b[t], ybuf[t], NN[t], F, F, 0, stream);
      layernorm_dual<<<(NN[t] * 32 + 255) / 256, 256, 0, stream>>>(
          ybuf[t], L[li].ln_g[t], L[li].ln_b[t], xcur[t], xs[t], SUMHID, jkoff, NN[t], F);
    }
    jkoff += F;
    in_dim = F;
  }

  // ---- JumpingKnowledge cat -> node MLP (gelu) ----
  for (int t = 0; t < 2; ++t)
    launch_gemm(xs[t], node_w[t], node_b[t], xfin[t], NN[t], SUMHID, 128, 1, stream);

  // ---- pooling into [G, 256] ----
  fill_f32<<<blks(NGRAPH * 256), 256, 0, stream>>>(hbuf[0], 0.0f, NGRAPH * 256);
  for (int t = 0; t < 2; ++t)
    pool_sum<<<blks(NN[t] * 128), 256, 0, stream>>>(xfin[t], batch[t], hbuf[0], NN[t], 128, t * 128, 256);

  // ---- graph MLP with eval-mode BN folded into scale ----
  const float bnscale = 1.0f / sqrtf(1.0f + 1e-5f);
  const int mdimK[3] = {256, 256, 128};
  const int mdimN[3] = {256, 128, 64};
  float* hin = hbuf[0];
  float* hout = hbuf[1];
  for (int i = 0; i < 3; ++i) {
    launch_gemm(hin, mlp_w[i], mlp_b[i], hout, NGRAPH, mdimK[i], mdimN[i], 0, stream);
    bn_gelu<<<blks(NGRAPH * mdimN[i]), 256, 0, stream>>>(hout, mlp_g[i], mlp_be[i], bnscale,
                                                         NGRAPH * mdimN[i], mdimN[i]);
    float* tmp = hin; hin = hout; hout = tmp;
  }
  final_out<<<1, 64, 0, stream>>>(hin, outw, outb, (float*)d_out, NGRAPH, 64);
}